// MixHop_49117245997550
// MI455X (gfx1250) — compile-verified
//
#include <hip/hip_runtime.h>
#include <math.h>

typedef __attribute__((ext_vector_type(2))) float v2f;
typedef __attribute__((ext_vector_type(8))) float v8f;

// ---------------- normalization kernels ----------------

__global__ void k_init_deg(float* __restrict__ deg, int n) {
    int i = blockIdx.x * blockDim.x + threadIdx.x;
    if (i < n) deg[i] = 1.0f;  // self-loop contributes 1 to every node's degree
}

__global__ void k_accum_deg(const int* __restrict__ dst, float* __restrict__ deg, int e) {
    int i = blockIdx.x * blockDim.x + threadIdx.x;
    if (i < e) atomicAdd(&deg[dst[i]], 1.0f);
}

__global__ void k_dinv(float* __restrict__ deg, int n) {
    int i = blockIdx.x * blockDim.x + threadIdx.x;
    if (i < n) {
        float d = deg[i];
        deg[i] = (d > 0.0f) ? rsqrtf(d) : 0.0f;  // in place: deg -> dinv
    }
}

__global__ void k_norm(const int* __restrict__ src, const int* __restrict__ dst,
                       const float* __restrict__ dinv, float* __restrict__ norm, int e) {
    int i = blockIdx.x * blockDim.x + threadIdx.x;
    if (i < e) norm[i] = dinv[src[i]] * dinv[dst[i]];
}

// ---------------- propagation (SpMM) ----------------

// out[i,:] = h[i,:] * dinv[i]^2   (the self-loop edge, also serves as zero-init)
__global__ void k_selfloop_init(const float* __restrict__ h, const float* __restrict__ dinv,
                                float* __restrict__ out, int n, int F) {
    int idx = blockIdx.x * blockDim.x + threadIdx.x;
    if (idx >= n * F) return;
    int i = idx / F;
    float d = dinv[i];
    out[idx] = h[idx] * (d * d);
}

// out[dst,:] += h[src,:] * norm[e]; one thread handles 4 features of one edge.
__global__ void k_scatter_edges(const float* __restrict__ h, const int* __restrict__ src,
                                const int* __restrict__ dst, const float* __restrict__ norm,
                                float* __restrict__ out, int e, int C /*F/4*/, int F) {
    int idx = blockIdx.x * blockDim.x + threadIdx.x;
    if (idx >= e * C) return;
    int ed = idx / C;
    int c  = idx % C;
    float nv = norm[ed];
    const float4* hp = (const float4*)(h + (size_t)src[ed] * F) + c;
    float4 v = *hp;
    float* op = out + (size_t)dst[ed] * F + (size_t)c * 4;
    atomicAdd(op + 0, v.x * nv);
    atomicAdd(op + 1, v.y * nv);
    atomicAdd(op + 2, v.z * nv);
    atomicAdd(op + 3, v.w * nv);
}

// ---------------- dense GEMM via fp32 WMMA ----------------
// out[:, colOff + 0..63] = H[N,K] @ W[K,64] + bias  (optional ReLU), out row stride = 192.
// One wave computes one 16x16 tile; 8 waves (256 thr) per block; N % 16 == 0 so no
// row bounds checks and EXEC is all-1s for every WMMA (ISA requirement).
__global__ void k_gemm_wmma(const float* __restrict__ H, const float* __restrict__ W,
                            const float* __restrict__ bias, float* __restrict__ out,
                            int K, int rowTiles, int colOff, int relu) {
    int wave = threadIdx.x >> 5;
    int lane = threadIdx.x & 31;
    int tile = blockIdx.x * 8 + wave;
    if (tile >= rowTiles * 4) return;          // whole wave exits uniformly
    int rt = tile >> 2, ct = tile & 3;
    int row0 = rt << 4, col0 = ct << 4;

    int mn = lane & 15;          // A: row-in-tile, B/C/D: col-in-tile
    int kb = (lane >> 4) << 1;   // K sub-offset: lanes 0-15 -> K+0,K+1; lanes 16-31 -> K+2,K+3

    const float* __restrict__ arow = H + (size_t)(row0 + mn) * K;
    const float* __restrict__ bcol = W + col0 + mn;

    v8f acc0 = {};
    v8f acc1 = {};
    for (int k = 0; k < K; k += 8) {
        v2f a0, b0, a1, b1;
        a0.x = arow[k + kb];
        a0.y = arow[k + kb + 1];
        b0.x = bcol[(size_t)(k + kb) * 64];
        b0.y = bcol[(size_t)(k + kb + 1) * 64];
        a1.x = arow[k + 4 + kb];
        a1.y = arow[k + 4 + kb + 1];
        b1.x = bcol[(size_t)(k + 4 + kb) * 64];
        b1.y = bcol[(size_t)(k + 4 + kb + 1) * 64];
        // D = A x B + C, fp32, 16x16x4
        acc0 = __builtin_amdgcn_wmma_f32_16x16x4_f32(false, a0, false, b0,
                                                     (short)0, acc0, false, false);
        acc1 = __builtin_amdgcn_wmma_f32_16x16x4_f32(false, a1, false, b1,
                                                     (short)0, acc1, false, false);
    }
    v8f acc = acc0 + acc1;

    float bv = bias[col0 + mn];
    int rbase = (lane >> 4) << 3;  // VGPR r: lanes 0-15 -> row r, lanes 16-31 -> row r+8
    float* orow = out + (size_t)(row0 + rbase) * 192 + colOff + col0 + mn;
#pragma unroll
    for (int r = 0; r < 8; ++r) {
        float v = acc[r] + bv;
        if (relu) v = fmaxf(v, 0.0f);
        orow[(size_t)r * 192] = v;
    }
}

// ---------------- launcher ----------------

extern "C" void kernel_launch(void* const* d_in, const int* in_sizes, int n_in,
                              void* d_out, int out_size, void* d_ws, size_t ws_size,
                              hipStream_t stream) {
    const float* x  = (const float*)d_in[0];   // [N,128]
    const int*   ei = (const int*)d_in[1];     // [2,E]
    const float* W1 = (const float*)d_in[2];   // [3,128,64]
    const float* b1 = (const float*)d_in[3];   // [3,64]
    const float* W2 = (const float*)d_in[4];   // [3,192,64]
    const float* b2 = (const float*)d_in[5];   // [3,64]
    float* out = (float*)d_out;                // [N,192]

    const int N = in_sizes[0] / 128;
    const int E = in_sizes[1] / 2;
    const int* src = ei;
    const int* dst = ei + E;

    // workspace layout (floats): dinv[N] | norm[E] | bufA[N*192] | bufB[N*192] | hcat[N*192]
    float* dinv = (float*)d_ws;
    float* norm = dinv + N;
    float* bufA = norm + E;
    float* bufB = bufA + (size_t)N * 192;
    float* hcat = bufB + (size_t)N * 192;

    const int T = 256;
    const int rowTiles = N / 16;                       // 3125
    const int gemmBlocks = (rowTiles * 4 + 7) / 8;     // 8 waves/block, 1 tile/wave

    // ---- gcn_norm ----
    k_init_deg <<<(N + T - 1) / T, T, 0, stream>>>(dinv, N);
    k_accum_deg<<<(E + T - 1) / T, T, 0, stream>>>(dst, dinv, E);
    k_dinv     <<<(N + T - 1) / T, T, 0, stream>>>(dinv, N);
    k_norm     <<<(E + T - 1) / T, T, 0, stream>>>(src, dst, dinv, norm, E);

    // ---- layer 1 propagations (F = 128) ----
    {
        const int F = 128, C = F / 4;
        k_selfloop_init<<<(N * F + T - 1) / T, T, 0, stream>>>(x, dinv, bufA, N, F);
        k_scatter_edges<<<(E * C + T - 1) / T, T, 0, stream>>>(x, src, dst, norm, bufA, E, C, F);
        k_selfloop_init<<<(N * F + T - 1) / T, T, 0, stream>>>(bufA, dinv, bufB, N, F);
        k_scatter_edges<<<(E * C + T - 1) / T, T, 0, stream>>>(bufA, src, dst, norm, bufB, E, C, F);
    }

    // ---- layer 1 GEMMs (+bias, ReLU) -> hcat[N,192] ----
    k_gemm_wmma<<<gemmBlocks, T, 0, stream>>>(x,    W1 + 0 * 128 * 64, b1 + 0,   hcat, 128, rowTiles, 0,   1);
    k_gemm_wmma<<<gemmBlocks, T, 0, stream>>>(bufA, W1 + 1 * 128 * 64, b1 + 64,  hcat, 128, rowTiles, 64,  1);
    k_gemm_wmma<<<gemmBlocks, T, 0, stream>>>(bufB, W1 + 2 * 128 * 64, b1 + 128, hcat, 128, rowTiles, 128, 1);

    // ---- layer 2 propagations (F = 192) ----
    {
        const int F = 192, C = F / 4;
        k_selfloop_init<<<(N * F + T - 1) / T, T, 0, stream>>>(hcat, dinv, bufA, N, F);
        k_scatter_edges<<<(E * C + T - 1) / T, T, 0, stream>>>(hcat, src, dst, norm, bufA, E, C, F);
        k_selfloop_init<<<(N * F + T - 1) / T, T, 0, stream>>>(bufA, dinv, bufB, N, F);
        k_scatter_edges<<<(E * C + T - 1) / T, T, 0, stream>>>(bufA, src, dst, norm, bufB, E, C, F);
    }

    // ---- layer 2 GEMMs (+bias) -> out[N,192] ----
    k_gemm_wmma<<<gemmBlocks, T, 0, stream>>>(hcat, W2 + 0 * 192 * 64, b2 + 0,   out, 192, rowTiles, 0,   0);
    k_gemm_wmma<<<gemmBlocks, T, 0, stream>>>(bufA, W2 + 1 * 192 * 64, b2 + 64,  out, 192, rowTiles, 64,  0);
    k_gemm_wmma<<<gemmBlocks, T, 0, stream>>>(bufB, W2 + 2 * 192 * 64, b2 + 128, out, 192, rowTiles, 128, 0);
}